// CapsuleLayer_4964982194646
// MI455X (gfx1250) — compile-verified
//
#include <hip/hip_runtime.h>
#include <hip/hip_bf16.h>

// ---------------------------------------------------------------------------
// Capsule layer with dynamic routing, MI455X (gfx1250, wave32).
//   u_hat  : f32 WMMA (v_wmma_f32_16x16x4_f32), stored bf16 in workspace
//   routing: 3 iterations of softmax -> weighted-sum+squash -> agreement
//            weighted-sum kernel stages u_hat chunks through LDS with
//            global_load_async_to_lds_b128 (ASYNCcnt domain).
// Workspace layout (bytes):
//   [0,          256MB)  U    : bf16  u_hat, layout [o][i][b][d] = [64][2048][32][32]
//   [256MB,      272MB)  bIJ  : f32   logits  [b][i][o]
//   [272MB,      288MB)  cIJ  : f32   softmax [b][i][o]
//   [288MB, 288MB+256KB) vbuf : f32   v_j     [b][o][d]
// ---------------------------------------------------------------------------

#define B_DIM    32
#define IN_CAPS  2048
#define IN_DIM   16
#define OUT_CAPS 64
#define OUT_DIM  32
#define M_DIM    (OUT_CAPS * OUT_DIM)   // 2048

typedef __attribute__((ext_vector_type(2))) float v2f;
typedef __attribute__((ext_vector_type(8))) float v8f;

// Pack two f32 -> packed bf16 pair, RNE, using only 32-bit integer ops
// (avoids v_mov_b16 half-register shuffling in codegen).
static __device__ __forceinline__ unsigned int pack_bf16_rne(float lo, float hi) {
    unsigned int a = __float_as_uint(lo);
    unsigned int b = __float_as_uint(hi);
    a += 0x7FFFu + ((a >> 16) & 1u);
    b += 0x7FFFu + ((b >> 16) & 1u);
    return (a >> 16) | (b & 0xFFFF0000u);
}

static __device__ __forceinline__ float2 bf16pair_to_f32(unsigned int u) {
    float2 r;
    r.x = __uint_as_float((u & 0x0000FFFFu) << 16);
    r.y = __uint_as_float(u & 0xFFFF0000u);
    return r;
}

// ---------------------------------------------------------------------------
// Kernel 1: u_hat via V_WMMA_F32_16X16X4_F32.
// One block per input capsule i. 8 waves: wave = (mt0 in 0..3, nt in 0..1).
// Each wave keeps its B-matrix (x) fragment in registers and loops M-tiles
// with pointer-stepped addressing (constant strides: W +4KB, U +8MB).
// A lane layout (16x4 f32):  row M = lane&15, k = 4*kk + 2*(lane>=16) + {0,1}
// B lane layout (4x16 f32):  col N = lane&15, k = 4*kk + 2*(lane>=16) + {0,1}
// D lane layout (16x16 f32): N = lane&15, M = r + 8*(lane>=16), r = 0..7
//   -> each lane's 8 outputs are consecutive in m = o*32+d; store as 16B bf16.
// ---------------------------------------------------------------------------
__global__ __launch_bounds__(256) void caps_uhat_wmma(
        const float* __restrict__ Wt,      // [IN_CAPS][M_DIM][IN_DIM]
        const float* __restrict__ X,       // [B][IN_CAPS][IN_DIM]
        unsigned short* __restrict__ U) {  // [OUT_CAPS][IN_CAPS][B][OUT_DIM] bf16
    const int i    = blockIdx.x;
    const int wav  = threadIdx.x >> 5;
    const int lane = threadIdx.x & 31;
    const int nt   = wav & 1;              // b-tile
    const int mt0  = wav >> 1;             // first m-tile
    const int half = lane >> 4;
    const int l16  = lane & 15;
    const int koff = 2 * half;

    const int bcol = nt * 16 + l16;        // batch column this lane owns (B & D)

    // B-matrix fragments for all 4 K-steps (reused across 32 M-tiles).
    const float* xrow = X + ((size_t)bcol * IN_CAPS + i) * IN_DIM;
    v2f bk[4];
#pragma unroll
    for (int kk = 0; kk < 4; ++kk) {
        const float2 t = *(const float2*)(xrow + kk * 4 + koff);
        bk[kk] = (v2f){t.x, t.y};
    }

    // Pointer-stepped A (W) read and U store addresses.
    const float* wptr =
        Wt + (size_t)i * (M_DIM * IN_DIM) + (size_t)(mt0 * 16 + l16) * IN_DIM + koff;
    const int m0 = mt0 * 16 + 8 * half;    // 0..56 -> o0 in {0,1}, dbase = m0&31
    unsigned short* uptr =
        U + ((((size_t)(m0 >> 5) * IN_CAPS + i) * B_DIM + bcol) * OUT_DIM + (m0 & 31));
    const size_t WSTEP = (size_t)64 * IN_DIM;                    // 64 rows
    const size_t USTEP = (size_t)2 * IN_CAPS * B_DIM * OUT_DIM;  // o += 2

    for (int it = 0; it < (M_DIM / 16) / 4; ++it) {
        __builtin_prefetch(wptr + WSTEP, 0, 0);   // next M-tile group

        v8f acc = {0.f, 0.f, 0.f, 0.f, 0.f, 0.f, 0.f, 0.f};
#pragma unroll
        for (int kk = 0; kk < 4; ++kk) {
            const float2 ta = *(const float2*)(wptr + kk * 4);
            v2f a = (v2f){ta.x, ta.y};
            acc = __builtin_amdgcn_wmma_f32_16x16x4_f32(
                    /*neg_a=*/false, a, /*neg_b=*/false, bk[kk],
                    /*c_mod=*/(short)0, acc,
                    /*reuse_a=*/false, /*reuse_b=*/false);
        }

        uint4 val = make_uint4(pack_bf16_rne(acc[0], acc[1]),
                               pack_bf16_rne(acc[2], acc[3]),
                               pack_bf16_rne(acc[4], acc[5]),
                               pack_bf16_rne(acc[6], acc[7]));
        *(uint4*)uptr = val;

        wptr += WSTEP;
        uptr += USTEP;
    }
}

// ---------------------------------------------------------------------------
// Kernel 2: c = softmax(b_ij) over o (64). One wave per (b,i) row.
// ---------------------------------------------------------------------------
__global__ __launch_bounds__(256) void caps_softmax(
        const float* __restrict__ bIJ, float* __restrict__ cIJ) {
    const size_t row  = (size_t)blockIdx.x * 8 + (threadIdx.x >> 5);  // b*IN_CAPS+i
    const int    lane = threadIdx.x & 31;

    const float2 v = *(const float2*)(bIJ + row * OUT_CAPS + lane * 2);
    float m = fmaxf(v.x, v.y);
#pragma unroll
    for (int s = 16; s >= 1; s >>= 1) m = fmaxf(m, __shfl_xor(m, s, 32));
    const float e0 = __expf(v.x - m);
    const float e1 = __expf(v.y - m);
    float sum = e0 + e1;
#pragma unroll
    for (int s = 16; s >= 1; s >>= 1) sum += __shfl_xor(sum, s, 32);
    const float inv = 1.0f / sum;
    float2 out = make_float2(e0 * inv, e1 * inv);
    *(float2*)(cIJ + row * OUT_CAPS + lane * 2) = out;
}

// ---------------------------------------------------------------------------
// Kernel 3: s_j[b,o,:] = sum_i c[b,i,o]*u_hat, then squash -> V[b,o,:].
// One block per (b,o). u_hat chunks (128 rows x 64B = 8KB) are staged into
// LDS with per-lane GLOBAL_LOAD_ASYNC_TO_LDS_B128 (ASYNCcnt), then reduced.
// ---------------------------------------------------------------------------
#define SJ_CHUNK 128   // input capsules per staged chunk

__global__ __launch_bounds__(256) void caps_sj_squash(
        const float* __restrict__ cIJ,
        const unsigned short* __restrict__ U,
        float* __restrict__ V) {
    const int b   = blockIdx.x >> 6;
    const int o   = blockIdx.x & 63;
    const int tid = threadIdx.x;
    const int d2  = tid & 15;
    const int ig  = tid >> 4;

    __shared__ __align__(16) unsigned int stage[SJ_CHUNK * 16];  // [i][d2], 8KB
    __shared__ float2 red[256];

    // Row i of this (b,o) lives at Ub + i*512 uints (2KB global row stride).
    const unsigned int* Ub =
        (const unsigned int*)U + ((size_t)o * IN_CAPS * B_DIM + b) * (OUT_DIM / 2);

    // Async staging pattern: thread t fetches 16B: row r = t>>2, quarter q = t&3;
    // issues rows r and r+64 (2 ops/lane/chunk).
    const int r = tid >> 2;
    const int q = tid & 3;
    const unsigned int  ldsA = (unsigned int)(size_t)&stage[r * 16 + q * 4];
    const unsigned int  ldsB = ldsA + 64 * 64;                       // rows +64
    const unsigned int* gBase = Ub + (size_t)r * 512 + q * 4;

    float2 acc = make_float2(0.f, 0.f);
    for (int chunk = 0; chunk < IN_CAPS; chunk += SJ_CHUNK) {
        const unsigned long long gA =
            (unsigned long long)(gBase + (size_t)chunk * 512);
        const unsigned long long gB = gA + (unsigned long long)64 * 512 * 4;
        asm volatile("global_load_async_to_lds_b128 %0, %1, off"
                     :: "v"(ldsA), "v"(gA) : "memory");
        asm volatile("global_load_async_to_lds_b128 %0, %1, off"
                     :: "v"(ldsB), "v"(gB) : "memory");
        asm volatile("s_wait_asynccnt 0x0" ::: "memory");
        __syncthreads();

        const float* cptr = cIJ + ((size_t)b * IN_CAPS + chunk) * OUT_CAPS + o;
        for (int l = ig; l < SJ_CHUNK; l += 16) {
            const float  c  = cptr[(size_t)l * OUT_CAPS];
            const float2 uv = bf16pair_to_f32(stage[l * 16 + d2]);
            acc.x += c * uv.x;
            acc.y += c * uv.y;
        }
        __syncthreads();   // protect stage[] before next chunk's async writes
    }

    red[tid] = acc;
    __syncthreads();
#pragma unroll
    for (int s = 8; s > 0; s >>= 1) {
        if (ig < s) {
            const float2 t = red[(ig + s) * 16 + d2];
            acc.x += t.x; acc.y += t.y;
            red[ig * 16 + d2] = acc;
        }
        __syncthreads();
    }

    if (tid < 16) {                         // lanes 0..15 of wave 0 hold s_j
        float n2 = acc.x * acc.x + acc.y * acc.y;
#pragma unroll
        for (int s = 8; s >= 1; s >>= 1) n2 += __shfl_xor(n2, s, 32);
        const float n     = sqrtf(n2);
        const float scale = n2 / ((1.f + n2) * (n + 1e-8f));
        float2 v = make_float2(acc.x * scale, acc.y * scale);
        *(float2*)(V + ((size_t)b * OUT_CAPS + o) * OUT_DIM + 2 * d2) = v;
    }
}

// ---------------------------------------------------------------------------
// Kernel 4: b_ij[b,i,o] += dot(u_hat[b,i,o,:], v[b,o,:]).
// One block per (b,o); each 16-lane half-wave handles one i per iteration.
// ---------------------------------------------------------------------------
__global__ __launch_bounds__(256) void caps_agree(
        const unsigned short* __restrict__ U,
        const float* __restrict__ V,
        float* __restrict__ bIJ) {
    const int b    = blockIdx.x >> 6;
    const int o    = blockIdx.x & 63;
    const int lane = threadIdx.x & 31;
    const int w    = threadIdx.x >> 5;
    const int d2   = lane & 15;
    const int ip   = lane >> 4;

    const float2 v =
        *(const float2*)(V + ((size_t)b * OUT_CAPS + o) * OUT_DIM + 2 * d2);
    const unsigned int* Ub = (const unsigned int*)U;

    for (int it = 0; it < IN_CAPS / 16; ++it) {
        const int i = it * 16 + w * 2 + ip;
        const unsigned int u =
            Ub[(((size_t)o * IN_CAPS + i) * B_DIM + b) * (OUT_DIM / 2) + d2];
        const float2 uv = bf16pair_to_f32(u);
        float dot = uv.x * v.x + uv.y * v.y;
#pragma unroll
        for (int s = 8; s >= 1; s >>= 1) dot += __shfl_xor(dot, s, 32);
        if (d2 == 0) {
            const size_t idx = ((size_t)b * IN_CAPS + i) * OUT_CAPS + o;
            bIJ[idx] += dot;                // unique writer per (b,i,o)
        }
    }
}

// ---------------------------------------------------------------------------
extern "C" void kernel_launch(void* const* d_in, const int* in_sizes, int n_in,
                              void* d_out, int out_size, void* d_ws, size_t ws_size,
                              hipStream_t stream) {
    (void)in_sizes; (void)n_in; (void)out_size; (void)ws_size;

    const float* X  = (const float*)d_in[0];   // [32, 2048, 16]
    const float* Wt = (const float*)d_in[1];   // [1, 2048, 64, 32, 16]

    char* ws = (char*)d_ws;
    const size_t U_BYTES   = (size_t)OUT_CAPS * IN_CAPS * B_DIM * OUT_DIM * 2; // 256 MB
    const size_t BIJ_BYTES = (size_t)B_DIM * IN_CAPS * OUT_CAPS * 4;           // 16 MB
    unsigned short* U    = (unsigned short*)ws;
    float*          bIJ  = (float*)(ws + U_BYTES);
    float*          cIJ  = (float*)(ws + U_BYTES + BIJ_BYTES);
    float*          vbuf = (float*)(ws + U_BYTES + 2 * BIJ_BYTES);

    hipMemsetAsync(bIJ, 0, BIJ_BYTES, stream);

    caps_uhat_wmma<<<IN_CAPS, 256, 0, stream>>>(Wt, X, U);

    const dim3 gridBO(B_DIM * OUT_CAPS);       // 2048 blocks = (b,o)
    const dim3 gridSM(B_DIM * IN_CAPS / 8);    // 8192 blocks, 8 rows each

    for (int t = 0; t < 3; ++t) {
        caps_softmax<<<gridSM, 256, 0, stream>>>(bIJ, cIJ);
        float* vdst = (t == 2) ? (float*)d_out : vbuf;
        caps_sj_squash<<<gridBO, 256, 0, stream>>>(cIJ, U, vdst);
        if (t < 2)
            caps_agree<<<gridBO, 256, 0, stream>>>(U, vbuf, bIJ);
    }
}